// Block_40080634806275
// MI455X (gfx1250) — compile-verified
//
#include <hip/hip_runtime.h>
#include <hip/hip_bf16.h>
#include <math.h>

typedef __attribute__((ext_vector_type(2))) float v2f;
typedef __attribute__((ext_vector_type(8))) float v8f;

#define BM 128
#define BN 64
#define BK 32
#define BKP 34  // padded LDS stride (floats), keeps 8B alignment & breaks bank conflicts

// --- CDNA5 async global->LDS copy (ASYNCcnt-tracked), inline asm per ISA 15.18.3 ---
__device__ __forceinline__ void async_lds_b128(unsigned lds_off, const float* gaddr) {
    asm volatile("global_load_async_to_lds_b128 %0, %1, off"
                 :: "v"(lds_off), "v"(gaddr) : "memory");
}
__device__ __forceinline__ void async_lds_b32(unsigned lds_off, const float* gaddr) {
    asm volatile("global_load_async_to_lds_b32 %0, %1, off"
                 :: "v"(lds_off), "v"(gaddr) : "memory");
}
__device__ __forceinline__ void wait_async_all() {
    asm volatile("s_wait_asynccnt 0x0" ::: "memory");
}
__device__ __forceinline__ unsigned lds_offset(const void* p) {
    // generic LDS pointer: low 32 bits are the offset within the workgroup LDS allocation
    return (unsigned)(size_t)p;
}

// C = act(A @ B + bias); A: MxK row-major (fp32), B: KxN (bmode=0 row-major,
// bmode=1 head-major (H,D,98) layout), C: MxN. V_WMMA_F32_16X16X4_F32 with
// async-double-buffered LDS tiles.
__global__ __launch_bounds__(256) void gemm_wmma_f32(
    const float* __restrict__ A, const float* __restrict__ B,
    const float* __restrict__ bias, float* __restrict__ C,
    int M, int N, int K, int bmode, int act)
{
    __shared__ float As[2][BM * BKP];   // [m][k]
    __shared__ float Bs[2][BN * BKP];   // [n][k] (transposed for b64 fragment loads)

    const int tid  = threadIdx.x;
    const int wave = tid >> 5;
    const int lane = tid & 31;
    const int half = lane >> 4;      // 0: lanes 0-15 (K pair 0,1) ; 1: lanes 16-31 (K pair 2,3)
    const int l    = lane & 15;

    const int wm = (wave & 3) * 32;  // 4 wave-rows
    const int wn = (wave >> 2) * 32; // 2 wave-cols
    const int m0 = blockIdx.y * BM;
    const int n0 = blockIdx.x * BN;

    // ---- precompute per-thread async-copy addresses (advance by constant per K-tile) ----
    unsigned aoff[4]; const float* agp[4];
    #pragma unroll
    for (int i = 0; i < 4; ++i) {
        int idx = tid + i * 256;
        int r  = idx >> 3;
        int c4 = (idx & 7) << 2;
        aoff[i] = lds_offset(&As[0][r * BKP + c4]);
        agp[i]  = A + (size_t)(m0 + r) * K + c4;   // 16B aligned (K % 4 == 0)
    }
    unsigned boff[8]; const float* bgp[8];
    #pragma unroll
    for (int i = 0; i < 8; ++i) {
        int idx = tid + i * 256;
        int k = idx >> 6;
        int n = idx & 63;
        int g = n0 + n;
        if (g > N - 1) g = N - 1;   // clamp: OOB columns only feed discarded C columns
        size_t off;
        if (bmode) {
            // W[h][d][e], h = g/98 via magic multiply (exact for g < 1568)
            int hh = (g * 669) >> 16;
            int e  = g - hh * 98;
            off = (size_t)hh * (size_t)K * 98u + (size_t)k * 98u + e;
        } else {
            off = (size_t)k * N + g;
        }
        boff[i] = lds_offset(&Bs[0][n * BKP + k]);
        bgp[i]  = B + off;
    }
    const size_t aAdv = BK;                                   // floats per K-tile (A)
    const size_t bAdv = bmode ? (size_t)BK * 98u : (size_t)BK * (size_t)N;
    const unsigned ABUF = (unsigned)(sizeof(float) * BM * BKP);
    const unsigned BBUF = (unsigned)(sizeof(float) * BN * BKP);

    v8f acc[2][2];
    acc[0][0] = {}; acc[0][1] = {}; acc[1][0] = {}; acc[1][1] = {};

    const int ntiles = K / BK;

    // prologue: stream tile 0 into buffer 0
    #pragma unroll
    for (int i = 0; i < 4; ++i) async_lds_b128(aoff[i], agp[i]);
    #pragma unroll
    for (int i = 0; i < 8; ++i) async_lds_b32(boff[i], bgp[i]);

    int buf = 0;
    for (int t = 0; t < ntiles; ++t) {
        wait_async_all();     // this wave's pending tile is in LDS
        __syncthreads();      // all waves done with previous compute + loads visible

        if (t + 1 < ntiles) { // stream next tile into the other buffer
            unsigned ab = (buf ^ 1) ? ABUF : 0u;
            unsigned bb = (buf ^ 1) ? BBUF : 0u;
            #pragma unroll
            for (int i = 0; i < 4; ++i) async_lds_b128(aoff[i] + ab, agp[i] + (size_t)(t + 1) * aAdv);
            #pragma unroll
            for (int i = 0; i < 8; ++i) async_lds_b32(boff[i] + bb, bgp[i] + (size_t)(t + 1) * bAdv);
        }

        const float* as = As[buf];
        const float* bs = Bs[buf];
        // ---- 8 x (4 WMMA) over the K-tile; fragments are ds_load_b64 pairs ----
        #pragma unroll
        for (int kk = 0; kk < BK; kk += 4) {
            v2f a0 = *(const v2f*)&as[(wm +      l) * BKP + kk + 2 * half];
            v2f a1 = *(const v2f*)&as[(wm + 16 + l) * BKP + kk + 2 * half];
            v2f b0 = *(const v2f*)&bs[(wn +      l) * BKP + kk + 2 * half];
            v2f b1 = *(const v2f*)&bs[(wn + 16 + l) * BKP + kk + 2 * half];
            acc[0][0] = __builtin_amdgcn_wmma_f32_16x16x4_f32(false, a0, false, b0, (short)0, acc[0][0], false, false);
            acc[0][1] = __builtin_amdgcn_wmma_f32_16x16x4_f32(false, a0, false, b1, (short)0, acc[0][1], false, false);
            acc[1][0] = __builtin_amdgcn_wmma_f32_16x16x4_f32(false, a1, false, b0, (short)0, acc[1][0], false, false);
            acc[1][1] = __builtin_amdgcn_wmma_f32_16x16x4_f32(false, a1, false, b1, (short)0, acc[1][1], false, false);
        }
        buf ^= 1;
    }

    // ---- epilogue: bias + optional exact GELU; C/D layout: VGPR r -> M = 8*half + r, N = l ----
    #pragma unroll
    for (int t = 0; t < 2; ++t) {
        #pragma unroll
        for (int u = 0; u < 2; ++u) {
            int col = n0 + wn + u * 16 + l;
            if (col < N) {
                float bv = bias ? bias[col] : 0.0f;
                #pragma unroll
                for (int r = 0; r < 8; ++r) {
                    int row = m0 + wm + t * 16 + half * 8 + r;
                    float v = acc[t][u][r] + bv;
                    if (act) v = 0.5f * v * (1.0f + erff(v * 0.70710678118654752f));
                    C[(size_t)row * N + col] = v;
                }
            }
        }
    }
}

__global__ __launch_bounds__(256) void layernorm_k(
    const float* __restrict__ x, const float* __restrict__ g,
    const float* __restrict__ b, float* __restrict__ y, int D)
{
    const int row = blockIdx.x;
    const float* xr = x + (size_t)row * D;
    float s = 0.0f, s2 = 0.0f;
    for (int i = threadIdx.x; i < D; i += 256) { float v = xr[i]; s += v; s2 += v * v; }
    #pragma unroll
    for (int off = 16; off > 0; off >>= 1) {
        s  += __shfl_xor(s,  off, 32);
        s2 += __shfl_xor(s2, off, 32);
    }
    __shared__ float rs[8], rs2[8];
    int wave = threadIdx.x >> 5, lane = threadIdx.x & 31;
    if (lane == 0) { rs[wave] = s; rs2[wave] = s2; }
    __syncthreads();
    if (wave == 0) {
        s  = (lane < 8) ? rs[lane]  : 0.0f;
        s2 = (lane < 8) ? rs2[lane] : 0.0f;
        #pragma unroll
        for (int off = 4; off > 0; off >>= 1) {
            s  += __shfl_xor(s,  off, 32);
            s2 += __shfl_xor(s2, off, 32);
        }
        if (lane == 0) { rs[0] = s; rs2[0] = s2; }
    }
    __syncthreads();
    s = rs[0]; s2 = rs2[0];
    float mu  = s / D;
    float var = s2 / D - mu * mu;
    float inv = rsqrtf(var + 1e-5f);
    for (int i = threadIdx.x; i < D; i += 256)
        y[(size_t)row * D + i] = (xr[i] - mu) * inv * g[i] + b[i];
}

// one block per (h, b): out_i = sum_j softmax_j(q_i * k_j * D^-0.5) * v_j, HS = 98
__global__ __launch_bounds__(128) void attn_k(
    const float* __restrict__ q, const float* __restrict__ k,
    const float* __restrict__ v, float* __restrict__ out)
{
    const int HS = 98, D = 1568;
    const int h = blockIdx.x;
    const int b = blockIdx.y;
    __shared__ float ks[98], vs[98];
    const size_t base = (size_t)b * D + (size_t)h * HS;
    int t = threadIdx.x;
    if (t < HS) { ks[t] = k[base + t]; vs[t] = v[base + t]; }
    __syncthreads();
    if (t < HS) {
        const float scale = 0.02525381361380526f;  // 1568^-0.5
        float qi = q[base + t] * scale;
        float m = -1e30f;
        for (int j = 0; j < HS; ++j) m = fmaxf(m, qi * ks[j]);
        float s = 0.0f, o = 0.0f;
        for (int j = 0; j < HS; ++j) {
            float w = expf(qi * ks[j] - m);
            s += w; o += w * vs[j];
        }
        out[base + t] = o / s;
    }
}

extern "C" void kernel_launch(void* const* d_in, const int* in_sizes, int n_in,
                              void* d_out, int out_size, void* d_ws, size_t ws_size,
                              hipStream_t stream)
{
    const float* x   = (const float*)d_in[0];
    const float* Wq  = (const float*)d_in[1];
    const float* Wk  = (const float*)d_in[2];
    const float* Wv  = (const float*)d_in[3];
    const float* Wo  = (const float*)d_in[4];
    const float* bo  = (const float*)d_in[5];
    const float* g1  = (const float*)d_in[6];
    const float* b1  = (const float*)d_in[7];
    const float* g2  = (const float*)d_in[8];
    const float* b2  = (const float*)d_in[9];
    const float* W1  = (const float*)d_in[10];
    const float* b1f = (const float*)d_in[11];
    const float* W2  = (const float*)d_in[12];
    const float* b2f = (const float*)d_in[13];

    const int Bn = 1024, D = 1568, FF = 6272, OUTD = 784;
    float* ws   = (float*)d_ws;
    float* h    = ws;
    float* q    = h    + (size_t)Bn * D;
    float* kbuf = q    + (size_t)Bn * D;
    float* vbuf = kbuf + (size_t)Bn * D;
    float* aout = vbuf + (size_t)Bn * D;
    float* proj = aout + (size_t)Bn * D;
    float* h2   = proj + (size_t)Bn * D;
    float* ff1  = h2   + (size_t)Bn * D;

    dim3 blk(256);
    dim3 gD((D + BN - 1) / BN, Bn / BM);

    layernorm_k<<<Bn, 256, 0, stream>>>(x, g1, b1, h, D);
    gemm_wmma_f32<<<gD, blk, 0, stream>>>(h, Wq, nullptr, q,    Bn, D, D, 1, 0);
    gemm_wmma_f32<<<gD, blk, 0, stream>>>(h, Wk, nullptr, kbuf, Bn, D, D, 1, 0);
    gemm_wmma_f32<<<gD, blk, 0, stream>>>(h, Wv, nullptr, vbuf, Bn, D, D, 1, 0);
    attn_k<<<dim3(16, Bn), 128, 0, stream>>>(q, kbuf, vbuf, aout);
    gemm_wmma_f32<<<gD, blk, 0, stream>>>(aout, Wo, bo, proj, Bn, D, D, 0, 0);
    layernorm_k<<<Bn, 256, 0, stream>>>(proj, g2, b2, h2, D);
    gemm_wmma_f32<<<dim3((FF + BN - 1) / BN, Bn / BM), blk, 0, stream>>>(h2, W1, b1f, ff1, Bn, FF, D, 0, 1);
    gemm_wmma_f32<<<dim3((OUTD + BN - 1) / BN, Bn / BM), blk, 0, stream>>>(ff1, W2, b2f, (float*)d_out, Bn, OUTD, FF, 0, 0);
}